// MaxMin2d_52802327937309
// MI455X (gfx1250) — compile-verified
//
#include <hip/hip_runtime.h>
#include <cstdint>

// ---------------------------------------------------------------------------
// MaxMin "conv" (max-min semiring im2col matmul) for MI455X / gfx1250.
//   out[b,f,h,w] = bias[f] + max_{c,kh,kw} min( xpad[b,c,h+kh-1,w+kw-1],
//                                               weight[(c*3+kh)*3+kw, f] )
// x[8,32,32,32] f32, weight[288,64] f32, bias[64] f32 -> out[8,64,32,32] f32.
//
// WMMA cannot express a (max,min) semiring -> compute core is VALU
// v_min_num_f32 / v_max3_num_f32. CDNA5-specific data movement:
//   * weights (73,728 B) DMA'd to LDS via Tensor Data Mover + s_wait_tensorcnt
//   * x tile staged via global_load_async_to_lds_b32 + s_wait_asynccnt
// ---------------------------------------------------------------------------

typedef uint32_t u32;
typedef uint64_t u64;
typedef __attribute__((ext_vector_type(4))) u32 v4u;
typedef __attribute__((ext_vector_type(8))) int v8i;
typedef __attribute__((ext_vector_type(4))) int v4i;

typedef __attribute__((address_space(1))) int gas_int;
typedef __attribute__((address_space(3))) int las_int;

#define C_IN   32
#define HW     32
#define F_OUT  64
#define K_TOT  288             // 32 * 3 * 3
#define WELEMS (K_TOT * F_OUT) // 18432 floats = 73728 B
#define ROWS_PB 4              // output rows per block
#define XROWS  6               // ROWS_PB + 2 halo
#define XCOLS  36              // 1 + 32 + 1 pad, rounded for 16B alignment

#if defined(__has_builtin)
#  if __has_builtin(__builtin_amdgcn_tensor_load_to_lds)
#    define HAVE_TDM 1
#  endif
#  if __has_builtin(__builtin_amdgcn_global_load_async_to_lds_b32)
#    define HAVE_ASYNC 1
#  endif
#endif
#ifndef HAVE_TDM
#  define HAVE_TDM 0
#endif
#ifndef HAVE_ASYNC
#  define HAVE_ASYNC 0
#endif

static __device__ inline void wait_asynccnt0() {
#if defined(__has_builtin) && __has_builtin(__builtin_amdgcn_s_wait_asynccnt)
    __builtin_amdgcn_s_wait_asynccnt(0);
#else
    asm volatile("s_wait_asynccnt 0" ::: "memory");
#endif
}

__global__ __launch_bounds__(256) void
MaxMin2d_52802327937309_kernel(const float* __restrict__ x,
                               const float* __restrict__ weight,
                               const float* __restrict__ bias,
                               float* __restrict__ out)
{
    __shared__ __align__(16) float wlds[WELEMS];               // 73,728 B
    __shared__ __align__(16) float xlds[C_IN * XROWS * XCOLS]; // 27,648 B

    const int tx   = threadIdx.x;        // 256 threads = 8 wave32
    const int tile = blockIdx.x;         // 0..7 : which 4-row band
    const int b    = blockIdx.y;         // 0..7 : batch
    const int h0   = tile * ROWS_PB;
    const int l0   = h0 * HW;            // 128 * tile

    // ---- stage weights into LDS via TDM ----------------------------------
#if HAVE_TDM
    if (tx < 32) {  // wave 0 issues one TDM descriptor (EXEC ignored by TDM)
        const u64 ga = (u64)(uintptr_t)weight;
        const u32 lo = (u32)(uintptr_t)(const void*)wlds; // low 32b = LDS addr

        v4u g0;
        g0[0] = 1u;                 // count=1, user descriptor, no gather
        g0[1] = lo;                 // lds_addr [63:32]
        g0[2] = (u32)ga;            // global_addr[31:0]
        g0[3] = (u32)((ga >> 32) & 0x01FFFFFFull) | (2u << 30); // addr hi + type=2

        v8i g1;
        g1[0] = (int)(2u << 16);                      // wg_mask=0, data_size=4B
        g1[1] = (int)((u32)(WELEMS & 0xFFFF) << 16);  // tensor_dim0[15:0]
        g1[2] = (int)(1u << 16);                      // dim0 hi=0, tensor_dim1=1
        g1[3] = (int)((u32)(WELEMS & 0xFFFF) << 16);  // tile_dim0 = 18432
        g1[4] = 0;                                    // tile_dim1/2 unused
        g1[5] = (int)WELEMS;                          // tensor_dim0_stride lo
        g1[6] = 0;
        g1[7] = 0;

        v4i gz4 = {0, 0, 0, 0};
        v8i gz8 = {0, 0, 0, 0, 0, 0, 0, 0};
        __builtin_amdgcn_tensor_load_to_lds(g0, g1, gz4, gz4, gz8, 0);
#  if defined(__has_builtin) && __has_builtin(__builtin_amdgcn_s_wait_tensorcnt)
        __builtin_amdgcn_s_wait_tensorcnt(0);
#  else
        asm volatile("s_wait_tensorcnt 0" ::: "memory");
#  endif
    }
#else
    for (int i = tx; i < WELEMS; i += 256) wlds[i] = weight[i];
#endif

    // ---- stage zero-padded x tile: 32c x 6rows x 36cols ------------------
    // col 0 == w=-1 pad, cols 1..32 == w, col 33 == w=32 pad, 34/35 filler.
#if HAVE_ASYNC
    // Pad cells via DS stores; interior cells ONLY via async-to-LDS (disjoint
    // byte ranges -> no DS/ASYNC ordering hazard before the barrier).
    for (int i = tx; i < C_IN * XROWS * XCOLS; i += 256) {
        const int r   = (i / XCOLS) % XROWS;
        const int col = i % XCOLS;
        const int h   = h0 + r - 1;
        if (!((unsigned)h < HW && col >= 1 && col <= HW))
            xlds[i] = 0.0f;
    }
    for (int i = tx; i < C_IN * XROWS * HW; i += 256) {
        const int c = i / (XROWS * HW);
        const int r = (i / HW) % XROWS;
        const int w = i % HW;
        const int h = h0 + r - 1;
        if ((unsigned)h < HW) {
            const float* gp = &x[(((b * C_IN + c) * HW) + h) * HW + w];
            const float* lp = &xlds[c * (XROWS * XCOLS) + r * XCOLS + (w + 1)];
            __builtin_amdgcn_global_load_async_to_lds_b32(
                (gas_int*)(uintptr_t)gp,
                (las_int*)(u32)(uintptr_t)lp, 0, 0);
        }
    }
    wait_asynccnt0();
#else
    for (int i = tx; i < C_IN * XROWS * XCOLS; i += 256) {
        const int c   = i / (XROWS * XCOLS);
        const int r   = (i / XCOLS) % XROWS;
        const int col = i % XCOLS;
        const int h   = h0 + r - 1;
        const int w   = col - 1;
        float v = 0.0f;
        if ((unsigned)h < HW && (unsigned)w < HW)
            v = x[(((b * C_IN + c) * HW) + h) * HW + w];
        xlds[i] = v;
    }
#endif
    __syncthreads();

    // ---- per-thread tile: 8 f x 4 contiguous w ---------------------------
    const int fg  = tx >> 5;            // 0..7
    const int lg  = tx & 31;            // 0..31
    const int row = lg >> 3;            // 0..3  (output row within band)
    const int wq  = (lg & 7) << 2;      // 0,4,...,28
    const int f0  = fg << 3;

    float acc[8][4];
#pragma unroll
    for (int fi = 0; fi < 8; ++fi)
#pragma unroll
        for (int li = 0; li < 4; ++li)
            acc[fi][li] = -__builtin_inff();

    for (int c = 0; c < C_IN; ++c) {
        const int cbase = c * (XROWS * XCOLS);
#pragma unroll
        for (int kh = 0; kh < 3; ++kh) {
            // 6-float window serves all three kw shifts for 4 outputs.
            const float* xr = &xlds[cbase + (row + kh) * XCOLS + wq];
            const float4 xa = *reinterpret_cast<const float4*>(xr);       // b128
            const float2 xb = *reinterpret_cast<const float2*>(xr + 4);   // b64
            const float pr[6] = {xa.x, xa.y, xa.z, xa.w, xb.x, xb.y};

            const int k0 = (c * 3 + kh) * 3;
            const float* wr0 = &wlds[(k0 + 0) * F_OUT + f0];
            const float* wr1 = &wlds[(k0 + 1) * F_OUT + f0];
            const float* wr2 = &wlds[(k0 + 2) * F_OUT + f0];
            const float4 a0 = *reinterpret_cast<const float4*>(wr0);
            const float4 b0 = *reinterpret_cast<const float4*>(wr0 + 4);
            const float4 a1 = *reinterpret_cast<const float4*>(wr1);
            const float4 b1 = *reinterpret_cast<const float4*>(wr1 + 4);
            const float4 a2 = *reinterpret_cast<const float4*>(wr2);
            const float4 b2 = *reinterpret_cast<const float4*>(wr2 + 4);
            const float w0[8] = {a0.x, a0.y, a0.z, a0.w, b0.x, b0.y, b0.z, b0.w};
            const float w1[8] = {a1.x, a1.y, a1.z, a1.w, b1.x, b1.y, b1.z, b1.w};
            const float w2[8] = {a2.x, a2.y, a2.z, a2.w, b2.x, b2.y, b2.z, b2.w};

#pragma unroll
            for (int fi = 0; fi < 8; ++fi)
#pragma unroll
                for (int li = 0; li < 4; ++li) {
                    // independent mins, then a max chain the backend can fuse
                    // into v_max3_num_f32.
                    const float m0 = __builtin_fminf(w0[fi], pr[li + 0]);
                    const float m1 = __builtin_fminf(w1[fi], pr[li + 1]);
                    const float m2 = __builtin_fminf(w2[fi], pr[li + 2]);
                    acc[fi][li] = __builtin_fmaxf(
                        __builtin_fmaxf(__builtin_fmaxf(acc[fi][li], m0), m1),
                        m2);
                }
        }
    }

    // ---- bias + store (coalesced b128 per f) -----------------------------
    const int lout = l0 + row * HW + wq;
#pragma unroll
    for (int fi = 0; fi < 8; ++fi) {
        const float bv = bias[f0 + fi];
        float4 o;
        o.x = acc[fi][0] + bv;
        o.y = acc[fi][1] + bv;
        o.z = acc[fi][2] + bv;
        o.w = acc[fi][3] + bv;
        *reinterpret_cast<float4*>(
            &out[((b * F_OUT + f0 + fi) << 10) + lout]) = o;
    }
}

extern "C" void kernel_launch(void* const* d_in, const int* in_sizes, int n_in,
                              void* d_out, int out_size, void* d_ws, size_t ws_size,
                              hipStream_t stream) {
    (void)in_sizes; (void)n_in; (void)out_size; (void)d_ws; (void)ws_size;
    const float* x      = (const float*)d_in[0];
    const float* weight = (const float*)d_in[1];
    const float* bias   = (const float*)d_in[2];
    float* out          = (float*)d_out;

    dim3 grid(8 /* row bands */, 8 /* batch */);
    MaxMin2d_52802327937309_kernel<<<grid, 256, 0, stream>>>(x, weight, bias, out);
}